// ST_GEN_74620761801552
// MI455X (gfx1250) — compile-verified
//
#include <hip/hip_runtime.h>
#include <hip/hip_bf16.h>
#include <math.h>

#define NN 2000
#define NE 24000
#define TS 64
#define TNROWS (TS * NN)  // 128000

typedef __attribute__((ext_vector_type(2))) float v2f;
typedef __attribute__((ext_vector_type(8))) float v8f;

enum { EPI_BIAS = 0, EPI_BIAS_RELU = 1, EPI_BIAS_BN_RELU = 2, EPI_ATOMIC = 3 };

// ---------------------------------------------------------------------------
// Generic fp32 WMMA GEMM: C = epilogue(A[MxK] @ B[KxN] (+bias)(+BN)(+relu))
// One 16x64 output tile per wave32 (4 N-subtiles): the A fragment is loaded
// once per k-step and reused by 4 independent V_WMMA_F32_16X16X4_F32 ops.
// kChunk > 0 enables split-K with fp32 atomic accumulation (EPI_ATOMIC).
// Requires M % 16 == 0, N % 64 == 0, K % 4 == 0 (and % kChunk for split-K).
// ---------------------------------------------------------------------------
template <int EPI>
__global__ __launch_bounds__(256) void gemm_wmma_f32(
    const float* __restrict__ A, const float* __restrict__ B,
    const float* __restrict__ bias, const float* __restrict__ gamma,
    const float* __restrict__ beta, const float* __restrict__ mean,
    const float* __restrict__ var, float* __restrict__ C, int M, int N, int K,
    int kChunk) {
  const int wavesPerBlock = blockDim.x >> 5;
  int wave = blockIdx.x * wavesPerBlock + ((int)threadIdx.x >> 5);
  int tilesN = N >> 6;  // 64-wide N tiles
  int totalTiles = (M >> 4) * tilesN;
  int nSplit = (kChunk > 0) ? (K / kChunk) : 1;
  if (wave >= totalTiles * nSplit) return;
  int tile = wave % totalTiles;
  int split = wave / totalTiles;
  int k0 = (kChunk > 0) ? split * kChunk : 0;
  int k1 = (kChunk > 0) ? (k0 + kChunk) : K;
  int tm = (tile / tilesN) << 4;
  int tn = (tile % tilesN) << 6;
  int lane = (int)threadIdx.x & 31;
  int r = lane & 15;
  int pair = lane >> 4;
  const float* Arow = A + (size_t)(tm + r) * (size_t)K;
  const float* Bcol = B + (size_t)(tn + r);
  v8f acc0 = {}, acc1 = {}, acc2 = {}, acc3 = {};
  for (int k = k0; k < k1; k += 4) {
    int ka = k + 2 * pair;
    v2f av;
    av.x = Arow[ka];
    av.y = Arow[ka + 1];
    const float* b0 = Bcol + (size_t)ka * (size_t)N;
    const float* b1 = Bcol + (size_t)(ka + 1) * (size_t)N;
    v2f bv0, bv1, bv2, bv3;
    bv0.x = b0[0];
    bv0.y = b1[0];
    bv1.x = b0[16];
    bv1.y = b1[16];
    bv2.x = b0[32];
    bv2.y = b1[32];
    bv3.x = b0[48];
    bv3.y = b1[48];
    acc0 = __builtin_amdgcn_wmma_f32_16x16x4_f32(false, av, false, bv0,
                                                 (short)0, acc0, false, false);
    acc1 = __builtin_amdgcn_wmma_f32_16x16x4_f32(false, av, false, bv1,
                                                 (short)0, acc1, false, false);
    acc2 = __builtin_amdgcn_wmma_f32_16x16x4_f32(false, av, false, bv2,
                                                 (short)0, acc2, false, false);
    acc3 = __builtin_amdgcn_wmma_f32_16x16x4_f32(false, av, false, bv3,
                                                 (short)0, acc3, false, false);
  }
  v8f accs[4] = {acc0, acc1, acc2, acc3};
#pragma unroll
  for (int s = 0; s < 4; ++s) {
    int col = tn + s * 16 + r;
    if (EPI == EPI_ATOMIC) {
#pragma unroll
      for (int rr = 0; rr < 8; ++rr) {
        int row = tm + rr + 8 * pair;
        atomicAdd(&C[(size_t)row * (size_t)N + col], accs[s][rr]);
      }
    } else {
      float bs = (bias != nullptr) ? bias[col] : 0.f;
      float g = 1.f, bt = 0.f, mu = 0.f, iv = 1.f;
      if (EPI == EPI_BIAS_BN_RELU) {
        g = gamma[col];
        bt = beta[col];
        mu = mean[col];
        iv = rsqrtf(var[col] + 1e-5f);
      }
#pragma unroll
      for (int rr = 0; rr < 8; ++rr) {
        int row = tm + rr + 8 * pair;
        float v = accs[s][rr] + bs;
        if (EPI == EPI_BIAS_BN_RELU) v = (v - mu) * iv * g + bt;
        if (EPI == EPI_BIAS_RELU || EPI == EPI_BIAS_BN_RELU) v = fmaxf(v, 0.f);
        C[(size_t)row * (size_t)N + col] = v;
      }
    }
  }
}

// ---------------------------------------------------------------------------
// gen1 input linear: (TN x 7) @ (7 x 64) + b  (K=7 too small for WMMA)
// ---------------------------------------------------------------------------
__global__ void lin1_kernel(const float* __restrict__ x,
                            const float* __restrict__ w,
                            const float* __restrict__ b,
                            float* __restrict__ out) {
  int id = blockIdx.x * blockDim.x + threadIdx.x;
  if (id >= TNROWS * 64) return;
  int c = id & 63;
  int tn = id >> 6;
  float acc = b[c];
#pragma unroll
  for (int i = 0; i < 7; ++i) acc += x[tn * 7 + i] * w[i * 64 + c];
  out[id] = acc;
}

// ---------------------------------------------------------------------------
// CSR build
// ---------------------------------------------------------------------------
__global__ void zero_i32(int* p, int n) {
  int i = blockIdx.x * blockDim.x + threadIdx.x;
  if (i < n) p[i] = 0;
}
__global__ void zero_f32(float* p, long n) {
  long i = (long)blockIdx.x * blockDim.x + threadIdx.x;
  if (i < n) p[i] = 0.f;
}
__global__ void deg_count(const int* __restrict__ dst, int* __restrict__ deg) {
  int e = blockIdx.x * blockDim.x + threadIdx.x;
  if (e < NE) atomicAdd(&deg[dst[e]], 1);
}
__global__ void scan_offsets(const int* __restrict__ deg, int* __restrict__ offs,
                             int* __restrict__ cursor) {
  if (blockIdx.x == 0 && threadIdx.x == 0) {
    int acc = 0;
    for (int n = 0; n < NN; ++n) {
      offs[n] = acc;
      cursor[n] = acc;
      acc += deg[n];
    }
    offs[NN] = acc;
  }
}
__global__ void fill_csr(const int* __restrict__ src, const int* __restrict__ dst,
                         int* __restrict__ cursor, int* __restrict__ esrc) {
  int e = blockIdx.x * blockDim.x + threadIdx.x;
  if (e < NE) {
    int p = atomicAdd(&cursor[dst[e]], 1);
    esrc[p] = src[e];
  }
}

// ---------------------------------------------------------------------------
// GENConv softmax aggregation: out = h + softmax-agg(relu(h[src])+eps) per dst
// block = (node,t), threads = channels
// ---------------------------------------------------------------------------
__global__ void gen_aggregate(const float* __restrict__ h, float* __restrict__ out,
                              const int* __restrict__ offs,
                              const int* __restrict__ esrc, int C) {
  int nt = blockIdx.x;
  int n = nt % NN;
  int t = nt / NN;
  int c = threadIdx.x;
  const float* ht = h + (size_t)t * NN * C;
  float hv = ht[(size_t)n * C + c];
  int s = offs[n], e = offs[n + 1];
  float mx = -3.4e38f;
  for (int i = s; i < e; ++i) {
    float m = fmaxf(ht[(size_t)esrc[i] * C + c], 0.f) + 1e-7f;
    mx = fmaxf(mx, m);
  }
  float den = 0.f, num = 0.f;
  for (int i = s; i < e; ++i) {
    float m = fmaxf(ht[(size_t)esrc[i] * C + c], 0.f) + 1e-7f;
    float ex = __expf(m - mx);
    den += ex;
    num += ex * m;
  }
  float agg = (e > s) ? num / (den + 1e-16f) : 0.f;
  out[(size_t)t * NN * C + (size_t)n * C + c] = hv + agg;
}

// ---------------------------------------------------------------------------
// CfC layer 1 (units=128, backbone=128).  xb = precomputed xt@bb_w[:256000].
// Persistent single block, 64 sequential steps, LDS-staged h and b.
// ---------------------------------------------------------------------------
__global__ __launch_bounds__(128) void cfc1_kernel(
    const float* __restrict__ xb, const float* __restrict__ bb_w,
    const float* __restrict__ bb_b, const float* __restrict__ ff1_w,
    const float* __restrict__ ff1_b, const float* __restrict__ ff2_w,
    const float* __restrict__ ff2_b, const float* __restrict__ ta_w,
    const float* __restrict__ ta_b, const float* __restrict__ tb_w,
    const float* __restrict__ tb_b, const float* __restrict__ proj_w,
    const float* __restrict__ proj_b, float* __restrict__ y) {
  __shared__ float h_s[128], b_s[128];
  int j = threadIdx.x;
  h_s[j] = 0.f;
  __syncthreads();
  const float* bbh = bb_w + (size_t)256000 * 128;  // recurrent rows of bb_w
  for (int t = 0; t < TS; ++t) {
    float acc = xb[t * 128 + j] + bb_b[j];
    for (int i = 0; i < 128; ++i) acc += h_s[i] * bbh[i * 128 + j];
    float bj = 1.7159f * tanhf(0.666f * acc);
    __syncthreads();
    b_s[j] = bj;
    __syncthreads();
    float a1 = ff1_b[j], a2 = ff2_b[j], a3 = ta_b[j] + tb_b[j];
    for (int i = 0; i < 128; ++i) {
      float bv = b_s[i];
      a1 += bv * ff1_w[i * 128 + j];
      a2 += bv * ff2_w[i * 128 + j];
      a3 += bv * (ta_w[i * 128 + j] + tb_w[i * 128 + j]);
    }
    float f1 = tanhf(a1), f2 = tanhf(a2);
    float ti = 1.f / (1.f + __expf(-a3));
    float hn = f1 * (1.f - ti) + ti * f2;
    __syncthreads();
    h_s[j] = hn;
    __syncthreads();
    float yo = proj_b[j];
    for (int i = 0; i < 128; ++i) yo += h_s[i] * proj_w[i * 128 + j];
    y[t * 128 + j] = yo;
  }
}

// ---------------------------------------------------------------------------
// CfC layer 2 (d_in=128, units=256, backbone=128)
// ---------------------------------------------------------------------------
__global__ __launch_bounds__(256) void cfc2_kernel(
    const float* __restrict__ yin, const float* __restrict__ bb_w,
    const float* __restrict__ bb_b, const float* __restrict__ ff1_w,
    const float* __restrict__ ff1_b, const float* __restrict__ ff2_w,
    const float* __restrict__ ff2_b, const float* __restrict__ ta_w,
    const float* __restrict__ ta_b, const float* __restrict__ tb_w,
    const float* __restrict__ tb_b, const float* __restrict__ proj_w,
    const float* __restrict__ proj_b, float* __restrict__ y) {
  __shared__ float x_s[128], h_s[256], b_s[128];
  int j = threadIdx.x;
  h_s[j] = 0.f;
  for (int t = 0; t < TS; ++t) {
    if (j < 128) x_s[j] = yin[t * 128 + j];
    __syncthreads();
    if (j < 128) {
      float acc = bb_b[j];
      for (int i = 0; i < 128; ++i) acc += x_s[i] * bb_w[i * 128 + j];
      for (int i = 0; i < 256; ++i) acc += h_s[i] * bb_w[(128 + i) * 128 + j];
      b_s[j] = 1.7159f * tanhf(0.666f * acc);
    }
    __syncthreads();
    float a1 = ff1_b[j], a2 = ff2_b[j], a3 = ta_b[j] + tb_b[j];
    for (int i = 0; i < 128; ++i) {
      float bv = b_s[i];
      a1 += bv * ff1_w[i * 256 + j];
      a2 += bv * ff2_w[i * 256 + j];
      a3 += bv * (ta_w[i * 256 + j] + tb_w[i * 256 + j]);
    }
    float f1 = tanhf(a1), f2 = tanhf(a2);
    float ti = 1.f / (1.f + __expf(-a3));
    float hn = f1 * (1.f - ti) + ti * f2;
    __syncthreads();
    h_s[j] = hn;
    __syncthreads();
    float yo = proj_b[j];
    for (int i = 0; i < 256; ++i) yo += h_s[i] * proj_w[i * 256 + j];
    y[t * 256 + j] = yo;
  }
}

// ---------------------------------------------------------------------------
static void launch_gemm(int epi, const float* A, const float* B, const float* bias,
                        const float* g, const float* bt, const float* mu,
                        const float* vr, float* C, int M, int N, int K,
                        int kChunk, hipStream_t s) {
  int tiles = (M >> 4) * (N >> 6);
  int nSplit = (kChunk > 0) ? (K / kChunk) : 1;
  int waves = tiles * nSplit;
  int blocks = (waves + 7) / 8;
  switch (epi) {
    case EPI_BIAS:
      gemm_wmma_f32<EPI_BIAS><<<blocks, 256, 0, s>>>(A, B, bias, g, bt, mu, vr, C, M, N, K, kChunk);
      break;
    case EPI_BIAS_RELU:
      gemm_wmma_f32<EPI_BIAS_RELU><<<blocks, 256, 0, s>>>(A, B, bias, g, bt, mu, vr, C, M, N, K, kChunk);
      break;
    case EPI_BIAS_BN_RELU:
      gemm_wmma_f32<EPI_BIAS_BN_RELU><<<blocks, 256, 0, s>>>(A, B, bias, g, bt, mu, vr, C, M, N, K, kChunk);
      break;
    default:
      gemm_wmma_f32<EPI_ATOMIC><<<blocks, 256, 0, s>>>(A, B, bias, g, bt, mu, vr, C, M, N, K, kChunk);
      break;
  }
}

extern "C" void kernel_launch(void* const* d_in, const int* in_sizes, int n_in,
                              void* d_out, int out_size, void* d_ws,
                              size_t ws_size, hipStream_t stream) {
  auto F = [&](int i) { return (const float*)d_in[i]; };
  // Flattened input order (recursive dict order of setup_inputs):
  // 0:x | gen1 1..10 | bn1 11..14 | gen2 15..24 | bn2 25..28 | gen3 29..36
  // bn3 37..40 | cfc1 41..52 | cfc2 53..64 | 65..68 fc | 69 edge_index
  const float* x = F(0);
  const int* ei = (const int*)d_in[69];
  const int* src = ei;
  const int* dst = ei + NE;
  float* out = (float*)d_out;

  // workspace carve
  char* w = (char*)d_ws;
  auto alloc = [&](size_t bytes) {
    void* p = (void*)w;
    w += (bytes + 255) & ~(size_t)255;
    return p;
  };
  float* bufA = (float*)alloc((size_t)TNROWS * 128 * 4);
  float* bufB = (float*)alloc((size_t)TNROWS * 128 * 4);
  float* bufU = (float*)alloc((size_t)TNROWS * 256 * 4);
  float* xb = (float*)alloc(TS * 128 * 4);
  float* y1 = (float*)alloc(TS * 128 * 4);
  float* y2 = (float*)alloc(TS * 256 * 4);
  float* fbuf = (float*)alloc(TS * 512 * 4);
  int* deg = (int*)alloc(NN * 4);
  int* offs = (int*)alloc((NN + 1) * 4);
  int* cursor = (int*)alloc(NN * 4);
  int* esrc = (int*)alloc(NE * 4);

  // ---- CSR build ----
  zero_i32<<<(NN + 255) / 256, 256, 0, stream>>>(deg, NN);
  deg_count<<<(NE + 255) / 256, 256, 0, stream>>>(dst, deg);
  scan_offsets<<<1, 1, 0, stream>>>(deg, offs, cursor);
  fill_csr<<<(NE + 255) / 256, 256, 0, stream>>>(src, dst, cursor, esrc);

  // ---- gen1 ----
  lin1_kernel<<<(TNROWS * 64 + 255) / 256, 256, 0, stream>>>(x, F(1), F(2), bufA);
  gen_aggregate<<<NN * TS, 64, 0, stream>>>(bufA, bufB, offs, esrc, 64);
  launch_gemm(EPI_BIAS_BN_RELU, bufB, F(3), F(4), F(5), F(6), F(7), F(8), bufU,
              TNROWS, 128, 64, 0, stream);
  launch_gemm(EPI_BIAS_BN_RELU, bufU, F(9), F(10), F(11), F(12), F(13), F(14),
              bufA, TNROWS, 64, 128, 0, stream);

  // ---- gen2 ----
  launch_gemm(EPI_BIAS, bufA, F(15), F(16), nullptr, nullptr, nullptr, nullptr,
              bufB, TNROWS, 128, 64, 0, stream);
  gen_aggregate<<<NN * TS, 128, 0, stream>>>(bufB, bufA, offs, esrc, 128);
  launch_gemm(EPI_BIAS_BN_RELU, bufA, F(17), F(18), F(19), F(20), F(21), F(22),
              bufU, TNROWS, 256, 128, 0, stream);
  launch_gemm(EPI_BIAS_BN_RELU, bufU, F(23), F(24), F(25), F(26), F(27), F(28),
              bufB, TNROWS, 128, 256, 0, stream);

  // ---- gen3 (no lin) ----
  gen_aggregate<<<NN * TS, 128, 0, stream>>>(bufB, bufA, offs, esrc, 128);
  launch_gemm(EPI_BIAS_BN_RELU, bufA, F(29), F(30), F(31), F(32), F(33), F(34),
              bufU, TNROWS, 256, 128, 0, stream);
  launch_gemm(EPI_BIAS_BN_RELU, bufU, F(35), F(36), F(37), F(38), F(39), F(40),
              bufB, TNROWS, 128, 256, 0, stream);
  // bufB is now seq: (64, 256000) row-major

  // ---- CfC1: hoisted input projection (split-K WMMA + f32 atomics) ----
  zero_f32<<<(TS * 128 + 255) / 256, 256, 0, stream>>>(xb, TS * 128);
  launch_gemm(EPI_ATOMIC, bufB, F(41), nullptr, nullptr, nullptr, nullptr,
              nullptr, xb, TS, 128, 256000, 1000, stream);
  cfc1_kernel<<<1, 128, 0, stream>>>(xb, F(41), F(42), F(43), F(44), F(45),
                                     F(46), F(47), F(48), F(49), F(50), F(51),
                                     F(52), y1);

  // ---- CfC2 ----
  cfc2_kernel<<<1, 256, 0, stream>>>(y1, F(53), F(54), F(55), F(56), F(57),
                                     F(58), F(59), F(60), F(61), F(62), F(63),
                                     F(64), y2);

  // ---- heads ----
  launch_gemm(EPI_BIAS_RELU, y2, F(65), F(66), nullptr, nullptr, nullptr,
              nullptr, fbuf, TS, 512, 256, 0, stream);
  launch_gemm(EPI_BIAS, fbuf, F(67), F(68), nullptr, nullptr, nullptr, nullptr,
              out, TS, 168000, 512, 0, stream);

  (void)in_sizes; (void)n_in; (void)out_size; (void)ws_size;
}